// CrownConv3d_76072460747257
// MI455X (gfx1250) — compile-verified
//
#include <hip/hip_runtime.h>
#include <hip/hip_bf16.h>

// CDNA5 / gfx1250 implicit-GEMM hexagonal conv3d.
//  - V_WMMA_F32_16X16X4_F32, fp32 end-to-end (matches reference numerics)
//  - each wave computes BOTH 16-OC m-tiles: B (input) ds_load shared across the
//    two WMMAs, A loads pair into one ds_load_2addr_b64, dual accumulator chains
//  - GLOBAL_LOAD_ASYNC_TO_LDS_B32 staging (ASYNCcnt), bank-conflict-free LDS
//  - non-temporal output stores (write-once 503 MB stream; preserve L2 for input)

typedef float v2f __attribute__((ext_vector_type(2)));
typedef float v8f __attribute__((ext_vector_type(8)));

#define IN_C   32
#define OUT_C  32
#define DEPTH  24

// taps t=0..20 live at dense positions p = (t/7)*9 + 1 + (t%7);
// widx maps tap -> index into (oc, ic, 21) weight tensor (W1_IDX / W2_IDX).
__constant__ int c_widx[2][21] = {
    { 5, 4, 0, 6, 3, 1, 2,   12, 11, 7, 13, 10, 8, 9,   19, 18, 14, 20, 17, 15, 16 },
    { 0, 5, 1, 6, 4, 2, 3,    7, 12, 8, 13, 11, 9, 10,  14, 19, 15, 20, 18, 16, 17 }
};

#define WLDS_F2   10752            // 21 taps * 8 chunks * 2 m * 32 lanes (float2) = 84 KB
#define ILDS_OFF  (WLDS_F2 * 2)    // float offset of input tile in dynamic LDS
#define ILDS_CPAD 34               // 32 channels + 2 pad -> conflict-free b64 reads

// Block: 8 waves cover NY y-rows x TX x-positions, all 32 OC.
// LDS input tile: 3*(NY+2) (z,y)-rows x (TX+2) x x 34 ch-padded.
template<int H, int W, int NY, int TX, int TAB>
__global__ __launch_bounds__(256)
void hexconv3d_wmma(const float* __restrict__ in,
                    const float* __restrict__ weight,
                    float* __restrict__ out)
{
    constexpr int XIW   = TX + 2;
    constexpr int YROWS = NY + 2;
    constexpr int RROWS = 3 * YROWS;
    constexpr int NXSUB = TX / 16;

    extern __shared__ float smem[];
    float2* wlds = reinterpret_cast<float2*>(smem);
    float*  ilds = smem + ILDS_OFF;

    const int tid = threadIdx.x;
    int bid = blockIdx.x;
    const int yblk = bid % (H / NY);      bid /= (H / NY);
    const int z    = bid % DEPTH;         bid /= DEPTH;
    const int b    = bid;
    const int ybase = yblk * NY;

    // ---- Stage weights into LDS in exact A-operand (16x4 f32) lane layout ----
    // wlds[((t*8 + chunk)*2 + m)*32 + lane] = {W[oc][c0], W[oc][c0+1]},
    // oc = m*16 + lane%16, c0 = chunk*4 + (lane<16 ? 0 : 2).
    for (int e = tid; e < WLDS_F2; e += 256) {
        const int lane  = e & 31;
        const int m     = (e >> 5) & 1;
        const int chunk = (e >> 6) & 7;
        const int t     = e >> 9;
        const int koff  = (lane >> 4) << 1;
        const int oc    = m * 16 + (lane & 15);
        const int c0    = chunk * 4 + koff;
        const int wi    = c_widx[TAB][t];
        float2 v;
        v.x = weight[(oc * IN_C + c0    ) * 21 + wi];
        v.y = weight[(oc * IN_C + c0 + 1) * 21 + wi];
        wlds[e] = v;
    }

    // ---- Stage input tile via GLOBAL_LOAD_ASYNC_TO_LDS_B32 (GVS mode) ----
    // ilds[(r*XIW + xi)*34 + c], r = dz*YROWS + dyy; replicate-pad via clamps.
    // Dynamic LDS segment starts at offset 0 -> LDS byte addr = f32 idx * 4.
    for (int e = tid; e < RROWS * XIW * IN_C; e += 256) {
        const int xi = e % XIW;
        const int c  = (e / XIW) & 31;
        const int r  = e / (XIW * IN_C);
        const int dz  = r / YROWS;
        const int dyy = r % YROWS;
        int zz = z - 1 + dz;      zz = zz < 0 ? 0 : (zz > DEPTH - 1 ? DEPTH - 1 : zz);
        int yy = ybase - 1 + dyy; yy = yy < 0 ? 0 : (yy > H - 1     ? H - 1     : yy);
        int xx = xi - 1;          xx = xx < 0 ? 0 : (xx > W - 1     ? W - 1     : xx);
        const size_t g = ((((size_t)b * IN_C + c) * DEPTH + zz) * H + yy) * (size_t)W + xx;
        const unsigned goff = (unsigned)(g * sizeof(float));
        const unsigned ldsb = (unsigned)((ILDS_OFF + (r * XIW + xi) * ILDS_CPAD + c) * sizeof(float));
        asm volatile("global_load_async_to_lds_b32 %0, %1, %2"
                     :: "v"(ldsb), "v"(goff), "s"(in) : "memory");
    }
    asm volatile("s_wait_asynccnt 0" ::: "memory");

    __syncthreads();

    // ---- WMMA compute: wave = (y-row, x-subtile); each wave does BOTH m-tiles
    const int lane   = tid & 31;
    const int wv     = tid >> 5;
    const int ry     = wv / NXSUB;
    const int xsub   = wv % NXSUB;
    const int ln     = lane & 15;
    const int koff   = (lane >> 4) << 1;       // 0 for lanes 0-15, 2 for 16-31
    const int xlocal = xsub * 16 + ln;

    v8f acc0 = {0.f, 0.f, 0.f, 0.f, 0.f, 0.f, 0.f, 0.f};
    v8f acc1 = {0.f, 0.f, 0.f, 0.f, 0.f, 0.f, 0.f, 0.f};

    const float*  ibase = ilds + (ry * XIW + xlocal) * ILDS_CPAD + koff;
    const float2* abase = wlds + lane;

    // Fully unrolled: tap geometry folds to DS immediate offsets. Per (t,chunk):
    // one shared B ds_load_b64 + one A ds_load_2addr_b64 feeding two WMMAs on
    // alternating accumulator chains (dependent-WMMA distance 2).
    #pragma unroll
    for (int t = 0; t < 21; ++t) {
        const int dz  = t / 7;
        const int q   = 1 + t % 7;
        const int dy  = q / 3;
        const int dxo = q % 3;
        const int r   = dz * YROWS + dy;
        const float* bb_t = ibase + (r * XIW + dxo) * ILDS_CPAD;
        #pragma unroll
        for (int chunk = 0; chunk < 8; ++chunk) {
            v2f bb = *reinterpret_cast<const v2f*>(bb_t + chunk * 4);
            v2f a0 = *reinterpret_cast<const v2f*>(abase + ((t * 8 + chunk) * 2 + 0) * 32);
            v2f a1 = *reinterpret_cast<const v2f*>(abase + ((t * 8 + chunk) * 2 + 1) * 32);
            acc0 = __builtin_amdgcn_wmma_f32_16x16x4_f32(
                false, a0, false, bb, (short)0, acc0, false, false);
            acc1 = __builtin_amdgcn_wmma_f32_16x16x4_f32(
                false, a1, false, bb, (short)0, acc1, false, false);
        }
    }

    // ---- Store: C/D layout -> VGPR rv holds M=rv (lanes 0-15), M=rv+8 (16-31)
    const int xg = xlocal;
    const int yg = ybase + ry;
    #pragma unroll
    for (int rv = 0; rv < 8; ++rv) {
        const int ochalf = rv + ((lane >> 4) << 3);
        const size_t g0 = ((((size_t)b * OUT_C + ochalf     ) * DEPTH + z) * H + yg) * (size_t)W + xg;
        const size_t g1 = ((((size_t)b * OUT_C + ochalf + 16) * DEPTH + z) * H + yg) * (size_t)W + xg;
        __builtin_nontemporal_store(acc0[rv], &out[g0]);
        __builtin_nontemporal_store(acc1[rv], &out[g1]);
    }
}

extern "C" void kernel_launch(void* const* d_in, const int* in_sizes, int n_in,
                              void* d_out, int out_size, void* d_ws, size_t ws_size,
                              hipStream_t stream) {
    const float* feat     = (const float*)d_in[0];
    const float* feat_row = (const float*)d_in[1];
    const float* weight   = (const float*)d_in[2];
    float* out = (float*)d_out;

    const int B = 10;
    const size_t n0 = (size_t)B * OUT_C * DEPTH * 64 * 128;

    // Tensor 0: (10,32,24,64,128): 1 y-row x 128 x per block, taps W1_IDX
    {
        constexpr int H = 64, W = 128, NY = 1, TX = 128;
        const size_t shmem = (ILDS_OFF + 3 * (NY + 2) * (TX + 2) * ILDS_CPAD) * sizeof(float);
        const int blocks = B * DEPTH * (H / NY) * (W / TX);
        hexconv3d_wmma<H, W, NY, TX, 0><<<blocks, 256, shmem, stream>>>(feat, weight, out);
    }
    // Tensor 1: (10,32,24,128,64): 2 y-rows x 64 x per block, taps W2_IDX
    {
        constexpr int H = 128, W = 64, NY = 2, TX = 64;
        const size_t shmem = (ILDS_OFF + 3 * (NY + 2) * (TX + 2) * ILDS_CPAD) * sizeof(float);
        const int blocks = B * DEPTH * (H / NY) * (W / TX);
        hexconv3d_wmma<H, W, NY, TX, 1><<<blocks, 256, shmem, stream>>>(feat_row, weight, out + n0);
    }
}